// TOYGNN_49357764165944
// MI455X (gfx1250) — compile-verified
//
#include <hip/hip_runtime.h>
#include <cstdint>

typedef float v2f __attribute__((ext_vector_type(2)));
typedef float v8f __attribute__((ext_vector_type(8)));

// ---------------------------------------------------------------------------
// Degree / normalization kernels
// ---------------------------------------------------------------------------
__global__ void k_init_deg(float* __restrict__ deg, int n) {
    int i = blockIdx.x * blockDim.x + threadIdx.x;
    if (i < n) deg[i] = 1.0f;  // self-loop contributes 1 to every node's degree
}

__global__ void k_deg_accum(const long long* __restrict__ dst,
                            float* __restrict__ deg, int e) {
    int i = blockIdx.x * blockDim.x + threadIdx.x;
    if (i < e) unsafeAtomicAdd(&deg[(int)dst[i]], 1.0f);
}

__global__ void k_rsqrt_inplace(float* __restrict__ deg, int n) {
    int i = blockIdx.x * blockDim.x + threadIdx.x;
    if (i < n) {
        float d = deg[i];
        deg[i] = (d > 0.0f) ? rsqrtf(d) : 0.0f;  // becomes dinv
    }
}

// ---------------------------------------------------------------------------
// Dense transform: out[N,DOUT] = (RELU? relu(in) : in)[N,K] @ W[K,DOUT]
// fp32 WMMA 16x16x4. One wave owns a 16x16 C tile; block covers 32 rows
// and all DOUT columns. W is staged transposed in LDS (row stride K+2 to
// break bank conflicts) so B fragments are contiguous 8B loads.
//
// ISA lane layouts (wave32):
//   A 16x4 : lane<16 -> {A[m][k+0],A[m][k+1]},  lane>=16 -> {A[m][k+2],A[m][k+3]}
//   B 4x16 : lane<16 -> {B[k+0][n],B[k+1][n]},  lane>=16 -> {B[k+2][n],B[k+3][n]}
//   C 16x16: vgpr r, lane<16 -> (row r,  col lane); lane>=16 -> (row r+8, col lane-16)
// ---------------------------------------------------------------------------
template <int K, int DOUT, bool RELU>
__global__ void __launch_bounds__(64 * (DOUT / 16))
k_gemm_wmma(const float* __restrict__ in, const float* __restrict__ W,
            float* __restrict__ out, int n_rows) {
    constexpr int KP = K + 2;                 // padded LDS row stride (floats)
    constexpr int CT = DOUT / 16;             // column tiles
    constexpr int NT = 64 * CT;               // threads per block
    __shared__ float Wt[DOUT * KP];           // W transposed: Wt[c][k]

    const int tid = threadIdx.x;
    for (int i = tid; i < K * DOUT; i += NT) {
        int k = i / DOUT;
        int c = i - k * DOUT;
        Wt[c * KP + k] = W[i];
    }
    __syncthreads();

    const int lane = tid & 31;
    const int wave = tid >> 5;
    const int row_tile = wave / CT;
    const int col_tile = wave - row_tile * CT;
    const int m    = lane & 15;
    const int koff = (lane >> 4) << 1;        // 0 for lanes 0-15, 2 for 16-31

    const long base_row = (long)blockIdx.x * 32 + row_tile * 16;
    if (base_row + 16 > n_rows) return;       // wave-uniform guard (EXEC stays all-1)

    const float* __restrict__ arow = in + (base_row + m) * K + koff;
    const float* brow = Wt + (col_tile * 16 + m) * KP + koff;

    v8f acc = {0.f, 0.f, 0.f, 0.f, 0.f, 0.f, 0.f, 0.f};
#pragma unroll
    for (int k = 0; k < K; k += 4) {
        v2f a = *(const v2f*)(arow + k);
        if (RELU) { a.x = fmaxf(a.x, 0.f); a.y = fmaxf(a.y, 0.f); }
        v2f b = *(const v2f*)(brow + k);
        acc = __builtin_amdgcn_wmma_f32_16x16x4_f32(
            /*neg_a=*/false, a, /*neg_b=*/false, b,
            /*c_mod=*/(short)0, acc, /*reuse_a=*/false, /*reuse_b=*/false);
    }

    const int col   = col_tile * 16 + m;
    const long rb   = base_row + ((lane >> 4) << 3);  // +8 rows for upper half-wave
#pragma unroll
    for (int r = 0; r < 8; ++r)
        out[(rb + r) * DOUT + col] = acc[r];
}

// ---------------------------------------------------------------------------
// acc[i][c] = bias[c] + t[i][c] * dinv[i]^2   (self-loop term folded in)
// ---------------------------------------------------------------------------
template <int DOUT>
__global__ void k_agg_init(const float* __restrict__ t, const float* __restrict__ dinv,
                           const float* __restrict__ bias, float* __restrict__ acc, int n) {
    long i = (long)blockIdx.x * blockDim.x + threadIdx.x;
    if (i >= (long)n * DOUT) return;
    int node = (int)(i / DOUT);
    int c    = (int)(i - (long)node * DOUT);
    float d  = dinv[node];
    acc[i] = bias[c] + t[i] * d * d;
}

// ---------------------------------------------------------------------------
// Edge scatter: acc[dst] += t[src] * dinv[src]*dinv[dst]
// DOUT/4 threads per edge; float4 gather (L2-resident) + 4 fp32 HW atomics.
// ---------------------------------------------------------------------------
template <int DOUT>
__global__ void k_agg_edges(const long long* __restrict__ src,
                            const long long* __restrict__ dst,
                            const float* __restrict__ dinv,
                            const float* __restrict__ t,
                            float* __restrict__ acc, int e) {
    constexpr int TPE = DOUT / 4;
    int tid   = blockIdx.x * blockDim.x + threadIdx.x;
    int edge  = tid / TPE;
    int chunk = tid - edge * TPE;
    if (edge >= e) return;
    int s = (int)src[edge];
    int d = (int)dst[edge];
    float w = dinv[s] * dinv[d];
    const float4 v = *(const float4*)(t + (long)s * DOUT + chunk * 4);
    float* p = acc + (long)d * DOUT + chunk * 4;
    unsafeAtomicAdd(p + 0, v.x * w);
    unsafeAtomicAdd(p + 1, v.y * w);
    unsafeAtomicAdd(p + 2, v.z * w);
    unsafeAtomicAdd(p + 3, v.w * w);
}

// ---------------------------------------------------------------------------
// Driver
// ---------------------------------------------------------------------------
extern "C" void kernel_launch(void* const* d_in, const int* in_sizes, int n_in,
                              void* d_out, int out_size, void* d_ws, size_t ws_size,
                              hipStream_t stream) {
    const float*     x   = (const float*)d_in[0];
    const long long* ei  = (const long long*)d_in[1];  // int64 [2, E]
    const float*     W1  = (const float*)d_in[2];
    const float*     b1  = (const float*)d_in[3];
    const float*     W2  = (const float*)d_in[4];
    const float*     b2  = (const float*)d_in[5];
    const float*     W3  = (const float*)d_in[6];
    const float*     b3  = (const float*)d_in[7];

    const int N = in_sizes[0] / 128;
    const int E = in_sizes[1] / 2;
    const long long* esrc = ei;
    const long long* edst = ei + E;

    // Workspace layout (needs ~60 MB):
    //   [0, 1MB)      : dinv (N floats)
    //   [1MB, 33MB)   : bufA (N*64 floats = 25.6 MB)
    //   [33MB, ...)   : bufB (N*64 floats = 25.6 MB)
    char* ws = (char*)d_ws;
    float* dinv = (float*)ws;
    float* bufA = (float*)(ws + ((size_t)1  << 20));
    float* bufB = (float*)(ws + ((size_t)33 << 20));
    float* outp = (float*)d_out;

    const int T = 256;

    // --- normalization: deg -> dinv ---
    k_init_deg  <<<(N + T - 1) / T, T, 0, stream>>>(dinv, N);
    k_deg_accum <<<(E + T - 1) / T, T, 0, stream>>>(edst, dinv, E);
    k_rsqrt_inplace<<<(N + T - 1) / T, T, 0, stream>>>(dinv, N);

    const int gemm_blocks = (N + 31) / 32;

    // --- layer 1: x[N,128] @ W1[128,64] -> aggregate -> bufB ---
    k_gemm_wmma<128, 64, false><<<gemm_blocks, 256, 0, stream>>>(x, W1, bufA, N);
    k_agg_init<64><<<(int)(((long)N * 64 + T - 1) / T), T, 0, stream>>>(bufA, dinv, b1, bufB, N);
    k_agg_edges<64><<<(int)(((long)E * 16 + T - 1) / T), T, 0, stream>>>(esrc, edst, dinv, bufA, bufB, E);

    // --- layer 2: relu(bufB)[N,64] @ W2[64,64] -> aggregate -> bufB ---
    k_gemm_wmma<64, 64, true><<<gemm_blocks, 256, 0, stream>>>(bufB, W2, bufA, N);
    k_agg_init<64><<<(int)(((long)N * 64 + T - 1) / T), T, 0, stream>>>(bufA, dinv, b2, bufB, N);
    k_agg_edges<64><<<(int)(((long)E * 16 + T - 1) / T), T, 0, stream>>>(esrc, edst, dinv, bufA, bufB, E);

    // --- layer 3: relu(bufB)[N,64] @ W3[64,32] -> aggregate -> d_out ---
    k_gemm_wmma<64, 32, true><<<gemm_blocks, 128, 0, stream>>>(bufB, W3, bufA, N);
    k_agg_init<32><<<(int)(((long)N * 32 + T - 1) / T), T, 0, stream>>>(bufA, dinv, b3, outp, N);
    k_agg_edges<32><<<(int)(((long)E * 8 + T - 1) / T), T, 0, stream>>>(esrc, edst, dinv, bufA, outp, E);
}